// PTGCN_2757369004686
// MI455X (gfx1250) — compile-verified
//
#include <hip/hip_runtime.h>

typedef __attribute__((ext_vector_type(2))) float v2f;
typedef __attribute__((ext_vector_type(8))) float v8f;

#define IN_CH 131
#define OUT_CH 64
#define FP_DIM 2065
#define HID 128
#define N_CLASSES 2
#define ZDIM (OUT_CH + FP_DIM) /* 2129 */

// ---------------- WMMA fragment helpers (f32 16x16x4) ----------------
// A 16x4 (MxK), row-major A[M][lda]: lane L holds row M=L&15,
// VGPR0 = K = k0 + 2*(L>=16), VGPR1 = K+1.   (predicate-free fast path)
__device__ __forceinline__ v2f ldA2(const float* __restrict__ A, int lda,
                                    int m0, int k0, int lane) {
  long long row = m0 + (lane & 15);
  int kb = k0 + ((lane >> 4) << 1);
  const float* p = A + row * lda + kb;
  v2f a;
  a.x = p[0];
  a.y = p[1];
  return a;
}
// tail: clamp address in-bounds, select result with 0 (no divergent branch)
__device__ __forceinline__ v2f ldA_t(const float* __restrict__ A, int lda,
                                     int m0, int k0, int lane, int kmax) {
  long long row = m0 + (lane & 15);
  int kb = k0 + ((lane >> 4) << 1);
  int kc = kb < (kmax - 2) ? kb : (kmax - 2);  // kc, kc+1 always valid
  const float* p = A + row * lda + kc;
  float x0 = p[0], x1 = p[1];
  v2f a;
  a.x = (kb     < kmax) ? x0 : 0.0f;
  a.y = (kb + 1 < kmax) ? x1 : 0.0f;
  return a;
}
// B 4x16 (KxN), row-major B[K][ldb]: lane L holds col N=L&15,
// VGPR0 = K-row k0 + 2*(L>=16), VGPR1 = next K-row.
__device__ __forceinline__ v2f ldB2(const float* __restrict__ B, int ldb,
                                    int k0, int n0, int lane) {
  int col = n0 + (lane & 15);
  int kb = k0 + ((lane >> 4) << 1);
  v2f b;
  b.x = B[(long long)kb * ldb + col];
  b.y = B[(long long)(kb + 1) * ldb + col];
  return b;
}
__device__ __forceinline__ v2f ldB_t(const float* __restrict__ B, int ldb,
                                     int k0, int n0, int lane, int kmax) {
  int col = n0 + (lane & 15);
  int kb = k0 + ((lane >> 4) << 1);
  int kc = kb < (kmax - 2) ? kb : (kmax - 2);
  float x0 = B[(long long)kc * ldb + col];
  float x1 = B[(long long)(kc + 1) * ldb + col];
  v2f b;
  b.x = (kb     < kmax) ? x0 : 0.0f;
  b.y = (kb + 1 < kmax) ? x1 : 0.0f;
  return b;
}
__device__ __forceinline__ v2f ldBs(const float* Bs, int ldb, int k0, int n0, int lane) {
  int col = n0 + (lane & 15);
  int kb = k0 + ((lane >> 4) << 1);
  v2f b;
  b.x = Bs[kb * ldb + col];
  b.y = Bs[(kb + 1) * ldb + col];
  return b;
}

#define WMMA_F32(a, b, c) \
  __builtin_amdgcn_wmma_f32_16x16x4_f32(false, (a), false, (b), (short)0, (c), false, false)

// ---------------- elementwise / graph kernels ----------------
__global__ void fill_kernel(float* __restrict__ p, float v, long long n) {
  long long t = blockIdx.x * (long long)blockDim.x + threadIdx.x;
  if (t < n) p[t] = v;
}

__global__ void deg_edges_kernel(const long long* __restrict__ ei,
                                 float* __restrict__ deg, long long E) {
  long long t = blockIdx.x * (long long)blockDim.x + threadIdx.x;
  if (t >= E) return;
  long long d = ei[E + t];  // dst row of edge_index
  atomicAdd(&deg[d], 1.0f);
}

__global__ void rsqrt_kernel(float* __restrict__ p, long long n) {
  long long t = blockIdx.x * (long long)blockDim.x + threadIdx.x;
  if (t < n) p[t] = rsqrtf(p[t]);  // deg >= 1 always (self loops)
}

// acc[i][c] = h[i][c] * dinv[i]^2   (self-loop contribution)
__global__ void self_loop_kernel(const float* __restrict__ h,
                                 const float* __restrict__ dinv,
                                 float* __restrict__ acc, long long total) {
  long long t = blockIdx.x * (long long)blockDim.x + threadIdx.x;
  if (t >= total) return;
  float di = dinv[t >> 6];
  acc[t] = h[t] * di * di;
}

// one wave per edge; lane handles channels {lane, lane+32}
__global__ void scatter_kernel(const long long* __restrict__ ei,
                               const float* __restrict__ h,
                               const float* __restrict__ dinv,
                               float* __restrict__ acc, long long E) {
  long long w = (blockIdx.x * (long long)blockDim.x + threadIdx.x) >> 5;
  int lane = threadIdx.x & 31;
  if (w >= E) return;
  long long s = ei[w];
  long long d = ei[E + w];
  float nrm = dinv[s] * dinv[d];
  const float* hs = h + s * 64;
  float* ad = acc + d * 64;
  atomicAdd(&ad[lane],      hs[lane]      * nrm);
  atomicAdd(&ad[lane + 32], hs[lane + 32] * nrm);
}

__global__ void bias_act_kernel(const float* __restrict__ acc,
                                const float* __restrict__ bias,
                                float* __restrict__ outp, long long total, int do_relu) {
  long long t = blockIdx.x * (long long)blockDim.x + threadIdx.x;
  if (t >= total) return;
  float v = acc[t] + bias[t & 63];
  outp[t] = do_relu ? fmaxf(v, 0.0f) : v;
}

// one wave per node: accumulate graph sums + counts
__global__ void pool_kernel(const float* __restrict__ h2,
                            const long long* __restrict__ bidx,
                            float* __restrict__ psum, float* __restrict__ pcnt,
                            long long N) {
  long long w = (blockIdx.x * (long long)blockDim.x + threadIdx.x) >> 5;
  int lane = threadIdx.x & 31;
  if (w >= N) return;
  long long g = bidx[w];
  atomicAdd(&psum[g * 64 + lane],      h2[w * 64 + lane]);
  atomicAdd(&psum[g * 64 + lane + 32], h2[w * 64 + lane + 32]);
  if (lane == 0) atomicAdd(&pcnt[g], 1.0f);
}

// z = [ pooled_mean (64) | fingerprint (2065) ]
__global__ void build_z_kernel(const float* __restrict__ psum,
                               const float* __restrict__ pcnt,
                               const float* __restrict__ fp,
                               float* __restrict__ z, long long G) {
  long long t = blockIdx.x * (long long)blockDim.x + threadIdx.x;
  long long total = G * ZDIM;
  if (t >= total) return;
  long long g = t / ZDIM;
  int k = (int)(t - g * ZDIM);
  z[t] = (k < OUT_CH) ? psum[g * 64 + k] / fmaxf(pcnt[g], 1.0f)
                      : fp[g * FP_DIM + (k - OUT_CH)];
}

// tiny head: out[g][c] = hid[g,:] . Wl2[:,c] + bl2[c]
__global__ void mlp2_kernel(const float* __restrict__ hid,
                            const float* __restrict__ Wl2,
                            const float* __restrict__ bl2,
                            float* __restrict__ outp, long long G) {
  long long t = blockIdx.x * (long long)blockDim.x + threadIdx.x;
  if (t >= G * N_CLASSES) return;
  long long g = t / N_CLASSES;
  int c = (int)(t - g * N_CLASSES);
  float s = bl2[c];
#pragma unroll 8
  for (int k = 0; k < HID; ++k) s += hid[g * HID + k] * Wl2[k * N_CLASSES + c];
  outp[t] = s;
}

// ---------------- WMMA GEMM kernels ----------------
// H[N,64] = X[N,131] @ W1[131,64]   (one wave -> 16x64 tile)
__global__ void __launch_bounds__(256) gemm_x_w1(const float* __restrict__ X,
                                                 const float* __restrict__ W1,
                                                 float* __restrict__ H, int nTiles) {
  int wave = blockIdx.x * (blockDim.x >> 5) + (threadIdx.x >> 5);
  if (wave >= nTiles) return;
  int lane = threadIdx.x & 31;
  int m0 = wave * 16;
  v8f c0 = {}, c1 = {}, c2 = {}, c3 = {};
  // 32 full K-steps, no predication
#pragma unroll 4
  for (int k0 = 0; k0 < 128; k0 += 4) {
    v2f a  = ldA2(X, IN_CH, m0, k0, lane);
    v2f b0 = ldB2(W1, OUT_CH, k0, 0,  lane);
    v2f b1 = ldB2(W1, OUT_CH, k0, 16, lane);
    v2f b2 = ldB2(W1, OUT_CH, k0, 32, lane);
    v2f b3 = ldB2(W1, OUT_CH, k0, 48, lane);
    c0 = WMMA_F32(a, b0, c0);
    c1 = WMMA_F32(a, b1, c1);
    c2 = WMMA_F32(a, b2, c2);
    c3 = WMMA_F32(a, b3, c3);
  }
  {  // K tail: k = 128..130 (branch-free select against 0)
    v2f a  = ldA_t(X, IN_CH, m0, 128, lane, IN_CH);
    v2f b0 = ldB_t(W1, OUT_CH, 128, 0,  lane, IN_CH);
    v2f b1 = ldB_t(W1, OUT_CH, 128, 16, lane, IN_CH);
    v2f b2 = ldB_t(W1, OUT_CH, 128, 32, lane, IN_CH);
    v2f b3 = ldB_t(W1, OUT_CH, 128, 48, lane, IN_CH);
    c0 = WMMA_F32(a, b0, c0);
    c1 = WMMA_F32(a, b1, c1);
    c2 = WMMA_F32(a, b2, c2);
    c3 = WMMA_F32(a, b3, c3);
  }
  int col = lane & 15;
  int rb = m0 + ((lane >> 4) << 3);
#pragma unroll
  for (int v = 0; v < 8; ++v) {
    long long r = (long long)(rb + v) * OUT_CH;
    H[r + col]      = c0[v];
    H[r + 16 + col] = c1[v];
    H[r + 32 + col] = c2[v];
    H[r + 48 + col] = c3[v];
  }
}

// Hout[N,64] = Hin[N,64] @ W2[64,64], W2 staged in LDS
__global__ void __launch_bounds__(256) gemm_h_w2(const float* __restrict__ Hin,
                                                 const float* __restrict__ W2,
                                                 float* __restrict__ Hout, int nTiles) {
  __shared__ float sW[64 * 68];
  for (int i = threadIdx.x; i < 64 * 64; i += blockDim.x) {
    int r = i >> 6, c = i & 63;
    sW[r * 68 + c] = W2[i];
  }
  __syncthreads();
  int wave = blockIdx.x * (blockDim.x >> 5) + (threadIdx.x >> 5);
  if (wave >= nTiles) return;
  int lane = threadIdx.x & 31;
  int m0 = wave * 16;
  v8f c0 = {}, c1 = {}, c2 = {}, c3 = {};
#pragma unroll 4
  for (int k0 = 0; k0 < OUT_CH; k0 += 4) {
    v2f a  = ldA2(Hin, OUT_CH, m0, k0, lane);
    v2f b0 = ldBs(sW, 68, k0, 0,  lane);
    v2f b1 = ldBs(sW, 68, k0, 16, lane);
    v2f b2 = ldBs(sW, 68, k0, 32, lane);
    v2f b3 = ldBs(sW, 68, k0, 48, lane);
    c0 = WMMA_F32(a, b0, c0);
    c1 = WMMA_F32(a, b1, c1);
    c2 = WMMA_F32(a, b2, c2);
    c3 = WMMA_F32(a, b3, c3);
  }
  int col = lane & 15;
  int rb = m0 + ((lane >> 4) << 3);
#pragma unroll
  for (int v = 0; v < 8; ++v) {
    long long r = (long long)(rb + v) * OUT_CH;
    Hout[r + col]      = c0[v];
    Hout[r + 16 + col] = c1[v];
    Hout[r + 32 + col] = c2[v];
    Hout[r + 48 + col] = c3[v];
  }
}

// Hid[G,128] = relu(Z[G,2129] @ Wl1[2129,128] + bl1)
__global__ void __launch_bounds__(256) gemm_z_wl1(const float* __restrict__ Z,
                                                  const float* __restrict__ Wl1,
                                                  const float* __restrict__ bl1,
                                                  float* __restrict__ Hid, int nTiles) {
  int wave = blockIdx.x * (blockDim.x >> 5) + (threadIdx.x >> 5);
  if (wave >= nTiles) return;
  int lane = threadIdx.x & 31;
  int m0 = wave * 16;
  v8f acc[8] = {};
  // 532 full K-steps (2128 = 4*532), no predication
  for (int k0 = 0; k0 < 2128; k0 += 4) {
    if ((k0 & 63) == 0)  // stride-ahead prefetch of the A stream
      __builtin_prefetch(&Z[(long long)(m0 + (lane & 15)) * ZDIM + k0 + 64], 0, 0);
    v2f a = ldA2(Z, ZDIM, m0, k0, lane);
#pragma unroll
    for (int nt = 0; nt < 8; ++nt) {
      v2f b = ldB2(Wl1, HID, k0, nt * 16, lane);
      acc[nt] = WMMA_F32(a, b, acc[nt]);
    }
  }
  {  // K tail: single valid k = 2128
    v2f a = ldA_t(Z, ZDIM, m0, 2128, lane, ZDIM);
#pragma unroll
    for (int nt = 0; nt < 8; ++nt) {
      v2f b = ldB_t(Wl1, HID, 2128, nt * 16, lane, ZDIM);
      acc[nt] = WMMA_F32(a, b, acc[nt]);
    }
  }
  int col = lane & 15;
  int rb = m0 + ((lane >> 4) << 3);
#pragma unroll
  for (int nt = 0; nt < 8; ++nt) {
    float bv = bl1[nt * 16 + col];
#pragma unroll
    for (int v = 0; v < 8; ++v)
      Hid[(long long)(rb + v) * HID + nt * 16 + col] = fmaxf(acc[nt][v] + bv, 0.0f);
  }
}

// ---------------- launcher ----------------
extern "C" void kernel_launch(void* const* d_in, const int* in_sizes, int n_in,
                              void* d_out, int out_size, void* d_ws, size_t ws_size,
                              hipStream_t stream) {
  (void)n_in; (void)out_size; (void)ws_size;
  const float*     x    = (const float*)d_in[0];
  const long long* ei   = (const long long*)d_in[1];
  const long long* bidx = (const long long*)d_in[2];
  const float*     fpr  = (const float*)d_in[3];
  const float*     W1   = (const float*)d_in[4];
  const float*     b1   = (const float*)d_in[5];
  const float*     W2   = (const float*)d_in[6];
  const float*     b2   = (const float*)d_in[7];
  const float*     Wl1  = (const float*)d_in[8];
  const float*     bl1  = (const float*)d_in[9];
  const float*     Wl2  = (const float*)d_in[10];
  const float*     bl2  = (const float*)d_in[11];
  float*           out  = (float*)d_out;

  const long long N = in_sizes[0] / IN_CH;    // 500000
  const long long E = in_sizes[1] / 2;        // 2000000
  const long long G = in_sizes[3] / FP_DIM;   // 8192

  // workspace layout (floats), contiguous:
  float* deg  = (float*)d_ws;                       // N   (becomes dinv in place)
  float* bufA = deg + ((N + 63) & ~63LL);           // N*64
  float* bufB = bufA + N * 64;                      // N*64
  float* psum = bufB + N * 64;                      // G*64
  float* pcnt = psum + G * 64;                      // G   (contiguous after psum)
  float* zbuf = bufA;                               // reuse: G*ZDIM <= N*64
  float* hid  = bufB;                               // reuse: G*HID  <= N*64

  dim3 blk(256);
  auto gridFor = [](long long total) { return dim3((unsigned)((total + 255) / 256)); };
  const int nT = (int)(N / 16);   // 31250 node-row tiles
  const int gT = (int)(G / 16);   // 512 graph-row tiles

  // degrees & normalization
  fill_kernel<<<gridFor(N), blk, 0, stream>>>(deg, 1.0f, N);          // self loop
  deg_edges_kernel<<<gridFor(E), blk, 0, stream>>>(ei, deg, E);
  rsqrt_kernel<<<gridFor(N), blk, 0, stream>>>(deg, N);               // deg -> dinv

  // conv1: h = relu(scatter(X@W1) + b1)
  gemm_x_w1<<<dim3((nT + 7) / 8), blk, 0, stream>>>(x, W1, bufA, nT);
  self_loop_kernel<<<gridFor(N * 64), blk, 0, stream>>>(bufA, deg, bufB, N * 64);
  scatter_kernel<<<gridFor(E * 32), blk, 0, stream>>>(ei, bufA, deg, bufB, E);
  bias_act_kernel<<<gridFor(N * 64), blk, 0, stream>>>(bufB, b1, bufA, N * 64, 1);

  // conv2: h2 = scatter(h@W2) + b2
  gemm_h_w2<<<dim3((nT + 7) / 8), blk, 0, stream>>>(bufA, W2, bufB, nT);
  self_loop_kernel<<<gridFor(N * 64), blk, 0, stream>>>(bufB, deg, bufA, N * 64);
  scatter_kernel<<<gridFor(E * 32), blk, 0, stream>>>(ei, bufB, deg, bufA, E);
  bias_act_kernel<<<gridFor(N * 64), blk, 0, stream>>>(bufA, b2, bufB, N * 64, 0);

  // pooling + MLP head
  fill_kernel<<<gridFor(G * 64 + G), blk, 0, stream>>>(psum, 0.0f, G * 64 + G);
  pool_kernel<<<gridFor(N * 32), blk, 0, stream>>>(bufB, bidx, psum, pcnt, N);
  build_z_kernel<<<gridFor(G * ZDIM), blk, 0, stream>>>(psum, pcnt, fpr, zbuf, G);
  gemm_z_wl1<<<dim3((gT + 7) / 8), blk, 0, stream>>>(zbuf, Wl1, bl1, hid, gT);
  mlp2_kernel<<<gridFor(G * N_CLASSES), blk, 0, stream>>>(hid, Wl2, bl2, out, G);
}